// PartitionTCA_57810259804604
// MI455X (gfx1250) — compile-verified
//
#include <hip/hip_runtime.h>

typedef __attribute__((ext_vector_type(2))) float v2f;
typedef __attribute__((ext_vector_type(4))) float v4f;
typedef __attribute__((ext_vector_type(8))) float v8f;

namespace {
constexpr int DI = 128, DJ = 512, DK = 512;
constexpr int TI = 16, TJ = 16, TK = 16;
constexpr int NBJ = DJ / TJ;   // 32
constexpr int NBK = DK / TK;   // 32
constexpr int NBLOCKS = (DI / TI) * NBJ * NBK;  // 8192

__device__ inline v8f wmma_k8(v2f alo, v2f ahi, v2f blo, v2f bhi, v8f c) {
  // D = A(16x4) * B(4x16) + C, chained twice for K = rank = 8
  c = __builtin_amdgcn_wmma_f32_16x16x4_f32(false, alo, false, blo, (short)0, c,
                                            false, false);
  c = __builtin_amdgcn_wmma_f32_16x16x4_f32(false, ahi, false, bhi, (short)0, c,
                                            false, false);
  return c;
}

__device__ inline void lds_add(float* p, float v) {
  // Non-returning LDS float atomic add -> ds_add_f32 (no load/add/store RMW)
  (void)__hip_atomic_fetch_add(p, v, __ATOMIC_RELAXED,
                               __HIP_MEMORY_SCOPE_WORKGROUP);
}
}  // namespace

__global__ __launch_bounds__(128) void partition_tca_kernel(
    const float* __restrict__ v00, const float* __restrict__ v01,
    const float* __restrict__ v10, const float* __restrict__ v11,
    const float* __restrict__ v20, const float* __restrict__ v21,
    float* __restrict__ out) {
  __shared__ float acc[TI * TJ * TK];  // 16 KB tile accumulator

  const int tid  = threadIdx.x;
  const int lane = tid & 31;
  const int wave = tid >> 5;
  const int half = lane >> 4;  // 0: lanes 0-15, 1: lanes 16-31
  const int ln   = lane & 15;

  const int tile = blockIdx.x;
  const int kt = tile & (NBK - 1);
  const int jt = (tile >> 5) & (NBJ - 1);
  const int it = tile >> 10;
  const int i0 = it * TI, j0 = jt * TJ, k0 = kt * TK;

  // Which 2 ranks of the K=4 chunk this half-wave supplies (VGPR0/VGPR1 pair).
  const int rb = half * 2;

  // ---- A operands (invariant across slabs within a tile) ----
  // A(16x4 f32): lane ln holds row M=ln; VGPR0 = K rb+0, VGPR1 = K rb+1.
  v2f a0l, a0h, a1l, a1h, a2l, a2h;
  a0l.x = v00[(0 + rb + 0) * DI + i0 + ln];
  a0l.y = v00[(0 + rb + 1) * DI + i0 + ln];
  a0h.x = v00[(4 + rb + 0) * DI + i0 + ln];
  a0h.y = v00[(4 + rb + 1) * DI + i0 + ln];

  a1l.x = v10[(0 + rb + 0) * DJ + j0 + ln];
  a1l.y = v10[(0 + rb + 1) * DJ + j0 + ln];
  a1h.x = v10[(4 + rb + 0) * DJ + j0 + ln];
  a1h.y = v10[(4 + rb + 1) * DJ + j0 + ln];

  a2l.x = v20[(0 + rb + 0) * DK + k0 + ln];
  a2l.y = v20[(0 + rb + 1) * DK + k0 + ln];
  a2h.x = v20[(4 + rb + 0) * DK + k0 + ln];
  a2h.y = v20[(4 + rb + 1) * DK + k0 + ln];

  // ---- Term 0 (init): out[i,j,k] = sum_r v00[r,i] * v01[r,j,k]
  // Per j-slab: M = i (16), N = k (16), K = r (2 chunks of 4).
  for (int q = 0; q < 4; ++q) {
    const int js = wave * 4 + q;
    const int jg = j0 + js;
    v2f bl, bh;
    bl.x = v01[((0 + rb + 0) * (size_t)DJ + jg) * DK + k0 + ln];
    bl.y = v01[((0 + rb + 1) * (size_t)DJ + jg) * DK + k0 + ln];
    bh.x = v01[((4 + rb + 0) * (size_t)DJ + jg) * DK + k0 + ln];
    bh.y = v01[((4 + rb + 1) * (size_t)DJ + jg) * DK + k0 + ln];
    v8f d = {};
    d = wmma_k8(a0l, a0h, bl, bh, d);
#pragma unroll
    for (int v = 0; v < 8; ++v) {
      const int mi = v + half * 8;  // i within tile
      acc[(mi * TJ + js) * TK + ln] = d[v];
    }
  }
  __syncthreads();

  // ---- Terms 1+2 merged (same i-slab ownership per wave, atomic LDS adds):
  //   out[i,j,k] += sum_r v10[r,j] * v11[r,i,k]   (M = j, N = k)
  //   out[i,j,k] += sum_r v20[r,k] * v21[r,i,j]   (M = k, N = j)
  for (int q = 0; q < 4; ++q) {
    const int is = wave * 4 + q;
    const int ig = i0 + is;

    v2f b1l, b1h, b2l, b2h;
    b1l.x = v11[((0 + rb + 0) * (size_t)DI + ig) * DK + k0 + ln];
    b1l.y = v11[((0 + rb + 1) * (size_t)DI + ig) * DK + k0 + ln];
    b1h.x = v11[((4 + rb + 0) * (size_t)DI + ig) * DK + k0 + ln];
    b1h.y = v11[((4 + rb + 1) * (size_t)DI + ig) * DK + k0 + ln];

    b2l.x = v21[((0 + rb + 0) * (size_t)DI + ig) * DJ + j0 + ln];
    b2l.y = v21[((0 + rb + 1) * (size_t)DI + ig) * DJ + j0 + ln];
    b2h.x = v21[((4 + rb + 0) * (size_t)DI + ig) * DJ + j0 + ln];
    b2h.y = v21[((4 + rb + 1) * (size_t)DI + ig) * DJ + j0 + ln];

    v8f d1 = {};
    d1 = wmma_k8(a1l, a1h, b1l, b1h, d1);
    v8f d2 = {};
    d2 = wmma_k8(a2l, a2h, b2l, b2h, d2);

#pragma unroll
    for (int v = 0; v < 8; ++v) {
      const int m = v + half * 8;
      lds_add(&acc[(is * TJ + m) * TK + ln], d1[v]);   // j = m, k = ln
      lds_add(&acc[(is * TJ + ln) * TK + m], d2[v]);   // j = ln, k = m
    }
  }
  __syncthreads();

  // ---- Store: 4096 f32 per tile, b128-vectorized (store-BW bound).
  const v4f* accv = (const v4f*)acc;
#pragma unroll
  for (int t = 0; t < 4; ++t) {
    const int rh  = t * 128 + tid;  // 0..511 half-rows
    const int row = rh >> 1;        // 0..255 = i*16 + j
    const int h   = rh & 1;
    const int i   = row >> 4;
    const int j   = row & 15;
    v4f lo = accv[row * 4 + h * 2 + 0];
    v4f hi = accv[row * 4 + h * 2 + 1];
    float* dst = out + ((size_t)(i0 + i) * DJ + (j0 + j)) * DK + k0 + h * 8;
    *(v4f*)(dst + 0) = lo;
    *(v4f*)(dst + 4) = hi;
  }
}

extern "C" void kernel_launch(void* const* d_in, const int* in_sizes, int n_in,
                              void* d_out, int out_size, void* d_ws,
                              size_t ws_size, hipStream_t stream) {
  const float* v00 = (const float*)d_in[0];
  const float* v01 = (const float*)d_in[1];
  const float* v10 = (const float*)d_in[2];
  const float* v11 = (const float*)d_in[3];
  const float* v20 = (const float*)d_in[4];
  const float* v21 = (const float*)d_in[5];
  float* out = (float*)d_out;

  partition_tca_kernel<<<NBLOCKS, 128, 0, stream>>>(v00, v01, v10, v11, v20,
                                                    v21, out);
}